// BoundaryLoss_51780125720647
// MI455X (gfx1250) — compile-verified
//
#include <hip/hip_runtime.h>
#include <cstdint>
#include <cstddef>

#define Bn   8
#define Cn   4
#define Hn   256
#define Wn   256
#define NCLS 3
#define XT   16
#define YT   64
#define HP   (Hn + 4)                       // padded column length (floats)
#define NBLK ((Wn / XT) * (Hn / YT) * Bn)   // 512 blocks per class
#define HWn  (Hn * Wn)                      // 65536
#define BHWn (Bn * Hn * Wn)                 // 524288

typedef __attribute__((ext_vector_type(2))) float        v2f;
typedef __attribute__((ext_vector_type(8))) float        v8f;
typedef __attribute__((ext_vector_type(4))) unsigned int u32x4;
typedef __attribute__((ext_vector_type(4))) int          i32x4;
typedef __attribute__((ext_vector_type(8))) int          i32x8;

// ---------------------------------------------------------------------------
// LDS inclusive scans (Hillis-Steele) over 256 threads.
// ---------------------------------------------------------------------------
__device__ __forceinline__ int scanIncMax(int* sh, int i, int v) {
  sh[i] = v; __syncthreads();
  #pragma unroll
  for (int off = 1; off < Wn; off <<= 1) {
    int t = sh[i];
    if (i >= off) t = max(t, sh[i - off]);
    __syncthreads();
    sh[i] = t;
    __syncthreads();
  }
  int r = sh[i];
  __syncthreads();
  return r;
}

__device__ __forceinline__ int scanIncMin(int* sh, int i, int v) {
  sh[i] = v; __syncthreads();
  #pragma unroll
  for (int off = 1; off < Wn; off <<= 1) {
    int t = sh[i];
    if (i >= off) t = min(t, sh[i - off]);
    __syncthreads();
    sh[i] = t;
    __syncthreads();
  }
  int r = sh[i];
  __syncthreads();
  return r;
}

// ---------------------------------------------------------------------------
// Kernel A: per-row 1-D squared distances along W for pos/neg masks of `cls`.
// Matches the reference scans exactly (INF=512 convention):
//   f[x] = x - prefmax(pos ? x : -513)       (no seed -> x+513)
//   r[x] = sufmin_{x'>=x}(pos ? x' : 768) - x (no seed -> 513+(W-1-x))
//   g = min(f,r)^2
// ---------------------------------------------------------------------------
__global__ __launch_bounds__(Wn)
void rowScanKernel(const int* __restrict__ targets,
                   float* __restrict__ gpos, float* __restrict__ gneg,
                   int* __restrict__ counts, int cls) {
  const int h = blockIdx.x;
  const int b = blockIdx.y;
  const int x = threadIdx.x;
  const int rx = Wn - 1 - x;

  __shared__ int sh[Wn];

  const int  t    = targets[(b * Hn + h) * Wn + x];
  const bool pos  = (t == cls);
  const int  t2   = targets[(b * Hn + h) * Wn + rx];
  const bool pos2 = (t2 == cls);

  // ---- pos mask ----
  int seedL = scanIncMax(sh, x, pos ? x : -513);
  // suffix-min via reversed slots: slot x owns position rx
  int seedRrev = scanIncMin(sh, x, pos2 ? rx : (Wn - 1 + 513));
  int fdist = x - seedL;            // for position x
  int rdistAtRx = seedRrev - rx;    // for position rx
  sh[rx] = rdistAtRx; __syncthreads();
  int rdist = sh[x]; __syncthreads();
  int d1p = min(fdist, rdist);
  gpos[(b * Hn + h) * Wn + x] = (float)(d1p * d1p);

  // ---- neg mask ----
  seedL = scanIncMax(sh, x, pos ? -513 : x);
  seedRrev = scanIncMin(sh, x, pos2 ? (Wn - 1 + 513) : rx);
  fdist = x - seedL;
  rdistAtRx = seedRrev - rx;
  sh[rx] = rdistAtRx; __syncthreads();
  rdist = sh[x]; __syncthreads();
  int d1n = min(fdist, rdist);
  gneg[(b * Hn + h) * Wn + x] = (float)(d1n * d1n);

  // ---- pos-pixel count (for the `has` gate; integer atomics: deterministic)
  sh[x] = pos ? 1 : 0; __syncthreads();
  #pragma unroll
  for (int off = Wn / 2; off > 0; off >>= 1) {
    if (x < off) sh[x] += sh[x + off];
    __syncthreads();
  }
  if (x == 0) atomicAdd(&counts[cls - 1], sh[0]);
}

// ---------------------------------------------------------------------------
// Transpose: g[b][h][w] -> gT[b][w][h] so envelope columns are contiguous.
// Processes both pos and neg arrays per block (coalesced in and out).
// ---------------------------------------------------------------------------
__global__ __launch_bounds__(256)
void transposeKernel(const float* __restrict__ srcP, float* __restrict__ dstP,
                     const float* __restrict__ srcN, float* __restrict__ dstN) {
  __shared__ float tile[32][33];
  const int b  = blockIdx.z;
  const int w0 = blockIdx.x * 32;
  const int h0 = blockIdx.y * 32;
  const int tx = threadIdx.x & 31;
  const int ty = threadIdx.x >> 5;   // 0..7

  const float* srcs[2] = { srcP, srcN };
  float*       dsts[2] = { dstP, dstN };
  #pragma unroll
  for (int a = 0; a < 2; ++a) {
    const float* src = srcs[a] + (size_t)b * HWn;
    float*       dst = dsts[a] + (size_t)b * HWn;
    #pragma unroll
    for (int i = 0; i < 32; i += 8)
      tile[ty + i][tx] = src[(h0 + ty + i) * Wn + (w0 + tx)];
    __syncthreads();
    #pragma unroll
    for (int i = 0; i < 32; i += 8)
      dst[(w0 + ty + i) * Hn + (h0 + tx)] = tile[tx][ty + i];
    __syncthreads();
  }
}

// ---------------------------------------------------------------------------
// TDM 2-D tile load, global -> LDS, with optional LDS padding so the DMA
// deposits a bank-conflict-free [XT][Hn+4] layout directly.
// Descriptor per CDNA5 ISA §8 (D# groups 0/1).
// ---------------------------------------------------------------------------
#if __has_builtin(__builtin_amdgcn_tensor_load_to_lds)
#define HAVE_TDM 1
__device__ __forceinline__ void tdm_load_2d(const float* gsrc, unsigned ldsOff,
                                            int dim0, int dim1, int stride0,
                                            bool pad) {
  unsigned long long ga = (unsigned long long)(uintptr_t)gsrc;
  u32x4 g0;
  g0[0] = 1u;                                     // count=1, user descriptor
  g0[1] = ldsOff;                                 // lds_addr (bytes)
  g0[2] = (unsigned)(ga & 0xFFFFFFFFu);           // global_addr[31:0]
  g0[3] = (unsigned)((ga >> 32) & 0x01FFFFFFu)    // global_addr[56:32]
        | (2u << 30);                             // type = 2 ("image")
  unsigned dw0 = (2u << 16);                      // data_size = 4B
  if (pad)
    dw0 |= (1u << 20)                             // pad_enable
         | (7u << 22)                             // pad_interval: 256 dwords
         | (3u << 25);                            // pad_amount:   4 dwords
  i32x8 g1;
  g1[0] = (int)dw0;
  g1[1] = (int)((unsigned)dim0 << 16);            // tensor_dim0[15:0]
  g1[2] = (int)((unsigned)dim1 << 16);            // tensor_dim1[15:0]
  g1[3] = (int)((unsigned)dim0 << 16);            // tile_dim0
  g1[4] = dim1;                                   // tile_dim1 (tile_dim2=0)
  g1[5] = stride0;                                // tensor_dim0_stride[31:0]
  g1[6] = 0;
  g1[7] = 0;
  i32x4 z4 = {0, 0, 0, 0};
#if defined(__clang_major__) && __clang_major__ >= 23
  i32x8 z8 = {0, 0, 0, 0, 0, 0, 0, 0};
  __builtin_amdgcn_tensor_load_to_lds(g0, g1, z4, z4, z8, 0);
#else
  __builtin_amdgcn_tensor_load_to_lds(g0, g1, z4, z4, 0);
#endif
}
#else
#define HAVE_TDM 0
#endif

// ---------------------------------------------------------------------------
// Kernel B: exact column lower-envelope (min-plus) + softmax + dmap*prob.
// One block per (b, 16-col tile, 64-row tile). Column tiles of the transposed
// g arrays are DMA'd into padded LDS by the TDM; the hot loop reads candidate
// streams with ds_load_b128 (4 candidates/load, conflict-free banking).
// ---------------------------------------------------------------------------
__global__ __launch_bounds__(256)
void envelopeKernel(const float* __restrict__ gposT, const float* __restrict__ gnegT,
                    const float* __restrict__ preds, const int* __restrict__ targets,
                    float* __restrict__ partials, int cls) {
  const int xt  = blockIdx.x;          // 0..15
  const int yt  = blockIdx.y;          // 0..3
  const int b   = blockIdx.z;          // 0..7
  const int tid = threadIdx.x;
  const int xl  = tid & (XT - 1);      // column in tile
  const int yo  = tid >> 4;            // 0..15
  const int x0  = xt * XT;
  const int y0  = yt * YT;

  __shared__ float gpT[XT][HP];        // padded: bank-conflict-free b128 reads
  __shared__ float gnT[XT][HP];
  __shared__ float wsum[8];

  const float* gpTile = gposT + (size_t)b * HWn + (size_t)x0 * Hn;
  const float* gnTile = gnegT + (size_t)b * HWn + (size_t)x0 * Hn;

#if HAVE_TDM
  if (tid == 0) {
    tdm_load_2d(gpTile, (unsigned)(uintptr_t)&gpT[0][0], Hn, XT, Hn, true);
    tdm_load_2d(gnTile, (unsigned)(uintptr_t)&gnT[0][0], Hn, XT, Hn, true);
    __builtin_amdgcn_s_wait_tensorcnt(0);
  }
#else
  for (int i = tid; i < XT * Hn; i += 256) {
    const int c = i >> 8, y = i & (Hn - 1);
    gpT[c][y] = gpTile[c * Hn + y];
    gnT[c][y] = gnTile[c * Hn + y];
  }
#endif
  __syncthreads();

  // ---- min-plus lower envelope, 4 outputs (y0+yo+16k) per thread ----
  float d2p[4], d2n[4], dyv[4];
  #pragma unroll
  for (int k = 0; k < 4; ++k) {
    d2p[k] = 3.0e38f; d2n[k] = 3.0e38f;
    dyv[k] = (float)(y0 + yo + k * 16);
  }

  const float4* gp4 = reinterpret_cast<const float4*>(&gpT[xl][0]);
  const float4* gn4 = reinterpret_cast<const float4*>(&gnT[xl][0]);
  #pragma unroll 2
  for (int q = 0; q < Hn / 4; ++q) {
    const float4 gv = gp4[q];
    const float4 hv = gn4[q];
    const float ge[4] = { gv.x, gv.y, gv.z, gv.w };
    const float he[4] = { hv.x, hv.y, hv.z, hv.w };
    #pragma unroll
    for (int e = 0; e < 4; ++e) {
      #pragma unroll
      for (int k = 0; k < 4; ++k) {
        d2p[k] = fminf(d2p[k], fmaf(dyv[k], dyv[k], ge[e]));
        d2n[k] = fminf(d2n[k], fmaf(dyv[k], dyv[k], he[e]));
      }
      #pragma unroll
      for (int k = 0; k < 4; ++k) dyv[k] -= 1.0f;
    }
  }

  // ---- fused epilogue: softmax prob, dmap, product ----
  const int xg = x0 + xl;
  const int* trow = targets + (size_t)b * HWn;
  const float* pb = preds + (size_t)b * Cn * HWn;

  float accA = 0.0f, accB = 0.0f;
  #pragma unroll
  for (int k = 0; k < 4; ++k) {
    const int y = y0 + yo + k * 16;
    const int   t   = trow[y * Wn + xg];
    const bool  pos = (t == cls);
    const float p0 = pb[0 * HWn + y * Wn + xg];
    const float p1 = pb[1 * HWn + y * Wn + xg];
    const float p2 = pb[2 * HWn + y * Wn + xg];
    const float p3 = pb[3 * HWn + y * Wn + xg];
    const float m  = fmaxf(fmaxf(p0, p1), fmaxf(p2, p3));
    const float e0 = __expf(p0 - m), e1 = __expf(p1 - m);
    const float e2 = __expf(p2 - m), e3 = __expf(p3 - m);
    const float esum = e0 + e1 + e2 + e3;
    const float ec   = (cls == 1) ? e1 : ((cls == 2) ? e2 : e3);
    const float prob = ec / esum;
    // dmap = edt_pos*neg - (edt_neg - 1)*pos  (edt_pos==0 at pos pixels)
    const float dmap = pos ? (1.0f - sqrtf(d2n[k])) : sqrtf(d2p[k]);
    const float v = dmap * prob;
    if (k < 2) accA += v; else accB += v;
  }

  // ---- block reduction through the matrix pipe ----
  // Per wave: 64 thread-partials (2/lane) form a 16x4 f32 A-matrix; B = ones
  // gives row sums; wave total = lane0 + lane16 of per-lane D-register sums.
  float wavesum;
#if __has_builtin(__builtin_amdgcn_wmma_f32_16x16x4_f32)
  {
    v2f a; a[0] = accA; a[1] = accB;
    v2f bb; bb[0] = 1.0f; bb[1] = 1.0f;
    v8f c = {};
    v8f d = __builtin_amdgcn_wmma_f32_16x16x4_f32(false, a, false, bb,
                                                  (short)0, c, false, false);
    float s = d[0] + d[1] + d[2] + d[3] + d[4] + d[5] + d[6] + d[7];
    wavesum = __shfl(s, 0, 32) + __shfl(s, 16, 32);
  }
#else
  {
    float s = accA + accB;
    #pragma unroll
    for (int off = 16; off > 0; off >>= 1) s += __shfl_down(s, off, 32);
    wavesum = __shfl(s, 0, 32);
  }
#endif

  const int wid  = tid >> 5;
  const int lane = tid & 31;
  if (lane == 0) wsum[wid] = wavesum;
  __syncthreads();
  if (tid == 0) {
    float bs = 0.0f;
    #pragma unroll
    for (int i = 0; i < 8; ++i) bs += wsum[i];
    const int blk = (b * 4 + yt) * 16 + xt;            // 0..NBLK-1
    partials[(cls - 1) * NBLK + blk] = bs;
  }
}

// ---------------------------------------------------------------------------
// Init + final (fixed-order, deterministic) reduction.
// ---------------------------------------------------------------------------
__global__ void initKernel(int* __restrict__ counts, float* __restrict__ partials) {
  const int i = blockIdx.x * blockDim.x + threadIdx.x;
  if (i < NCLS) counts[i] = 0;
  if (i < NCLS * NBLK) partials[i] = 0.0f;
}

__global__ void finalKernel(const int* __restrict__ counts,
                            const float* __restrict__ partials,
                            float* __restrict__ out) {
  if (threadIdx.x == 0 && blockIdx.x == 0) {
    float total = 0.0f, cnt = 0.0f;
    for (int c = 0; c < NCLS; ++c) {
      float s = 0.0f;
      for (int i = 0; i < NBLK; ++i) s += partials[c * NBLK + i];
      const float loss = s / (float)BHWn;
      if (counts[c] > 0) { total += loss; cnt += 1.0f; }
    }
    out[0] = (cnt > 0.0f) ? (total / fmaxf(cnt, 1.0f)) : 0.0f;
  }
}

// ---------------------------------------------------------------------------
// Host launcher. Scratch: gpos | gneg | gposT | gnegT (2MB each) | counts |
// partials (~8.4MB total).
// ---------------------------------------------------------------------------
extern "C" void kernel_launch(void* const* d_in, const int* in_sizes, int n_in,
                              void* d_out, int out_size, void* d_ws, size_t ws_size,
                              hipStream_t stream) {
  (void)in_sizes; (void)n_in; (void)out_size; (void)ws_size;
  const float* preds   = (const float*)d_in[0];   // [8,4,256,256] f32
  const int*   targets = (const int*)d_in[1];     // [8,256,256] i32
  float* out = (float*)d_out;

  char* ws = (char*)d_ws;
  float* gpos     = (float*)(ws);
  float* gneg     = (float*)(ws + (size_t)BHWn * 4);
  float* gposT    = (float*)(ws + (size_t)BHWn * 8);
  float* gnegT    = (float*)(ws + (size_t)BHWn * 12);
  int*   counts   = (int*)  (ws + (size_t)BHWn * 16);
  float* partials = (float*)(ws + (size_t)BHWn * 16 + 256);

  initKernel<<<(NCLS * NBLK + 255) / 256, 256, 0, stream>>>(counts, partials);

  for (int cls = 1; cls <= NCLS; ++cls) {
    rowScanKernel<<<dim3(Hn, Bn), Wn, 0, stream>>>(targets, gpos, gneg, counts, cls);
    transposeKernel<<<dim3(Wn / 32, Hn / 32, Bn), 256, 0, stream>>>(
        gpos, gposT, gneg, gnegT);
    envelopeKernel<<<dim3(Wn / XT, Hn / YT, Bn), 256, 0, stream>>>(
        gposT, gnegT, preds, targets, partials, cls);
  }

  finalKernel<<<1, 32, 0, stream>>>(counts, partials, out);
}